// WraperStdcUpsample_36876589204064
// MI455X (gfx1250) — compile-verified
//
#include <hip/hip_runtime.h>

// WraperStdcUpsample: 8x nearest upsample -> channel softmax -> (max, argmax).
// Collapsed to per-low-res-pixel compute + 8x8 replicated coalesced b128 stores.
// Input tile staged into LDS with the CDNA5 Tensor Data Mover (TENSOR_LOAD_TO_LDS).

typedef __attribute__((ext_vector_type(4))) unsigned int u32x4;
typedef __attribute__((ext_vector_type(8))) int          i32x8;
typedef __attribute__((ext_vector_type(4))) int          i32x4;

#define B_ 2
#define C_ 19
#define H_ 128
#define W_ 256
#define S_ 8

__global__ void __launch_bounds__(W_) seg_upsample_softmax_argmax(
    const float* __restrict__ x,          // [B, C, H, W]
    float* __restrict__ score_out,        // [B, 8H, 8W] f32
    float* __restrict__ label_out)        // [B, 8H, 8W] (argmax, stored as f32)
{
    __shared__ float tile[C_ * W_];       // 19 x 256 f32 = 19 KB

    const int bh = blockIdx.x;            // one block per (b, h): 2*128 = 256 blocks
    const int b  = bh / H_;
    const int h  = bh - b * H_;
    const int t  = threadIdx.x;           // == w, 0..255

    // ---- TDM: DMA the 19x256 channel tile for this (b,h) into LDS ----------
    // Tile start = &x[b][0][h][0]; dim0 = w (256 contiguous f32),
    // dim1 = c with stride H*W = 32768 elements.
    if (t < 32) {  // wave 0 issues the DMA (uniform branch; EXEC ignored by TDM)
        unsigned long long gaddr =
            (unsigned long long)(const void*)(x + ((size_t)b * C_ * H_ + (size_t)h) * W_);
        unsigned int lds_addr =
            (unsigned int)(unsigned long long)(const void*)&tile[0];

        // D# group 0 (128b): count=1 | lds_addr | global_addr[56:0] | type=2
        u32x4 g0;
        g0[0] = 1u;                                   // count=1, is_restore=0, gather=0
        g0[1] = lds_addr;                             // LDS byte address
        g0[2] = (unsigned int)gaddr;                  // global_addr[31:0]
        g0[3] = (unsigned int)((gaddr >> 32) & 0x1FFFFFFull) | (2u << 30); // [56:32], type=2

        // D# group 1 (256b)
        i32x8 g1;
        g1[0] = (int)(2u << 16);        // wg_mask=0, data_size=2 (4 bytes), no flags
        g1[1] = (int)((unsigned)W_ << 16);  // tensor_dim0[15:0]=256 @ bits 63:48
        g1[2] = (int)((unsigned)C_ << 16);  // tensor_dim0[31:16]=0, tensor_dim1[15:0]=19
        g1[3] = (int)((unsigned)W_ << 16);  // tensor_dim1[31:16]=0, tile_dim0=256
        g1[4] = C_;                     // tile_dim1=19, tile_dim2=0
        g1[5] = H_ * W_;                // tensor_dim0_stride[31:0] = 32768 elements
        g1[6] = 0;                      // stride[47:32]=0, tensor_dim1_stride lo=0
        g1[7] = 0;

        i32x4 gz; gz[0] = 0; gz[1] = 0; gz[2] = 0; gz[3] = 0;   // 2-D: groups 2/3 unused
        i32x8 gz8; gz8[0]=0; gz8[1]=0; gz8[2]=0; gz8[3]=0;
        gz8[4]=0; gz8[5]=0; gz8[6]=0; gz8[7]=0;

        __builtin_amdgcn_tensor_load_to_lds(g0, g1, gz, gz, gz8, 0);
        __builtin_amdgcn_s_wait_tensorcnt(0);
    }
    __syncthreads();

    // ---- per-pixel channel softmax-max / argmax (registers, LDS conflict-free)
    float v[C_];
#pragma unroll
    for (int c = 0; c < C_; ++c) v[c] = tile[c * W_ + t];

    float m = v[0];
    int   idx = 0;
#pragma unroll
    for (int c = 1; c < C_; ++c)
        if (v[c] > m) { m = v[c]; idx = c; }   // strict '>' => first-max like jnp.argmax

    float s = 0.0f;
#pragma unroll
    for (int c = 0; c < C_; ++c) s += __expf(v[c] - m);

    const float score  = 1.0f / s;             // exp(m-m)/sum = max of softmax
    const float flabel = (float)idx;

    // ---- write the 8x8 patch: two b128 stores per row per output ------------
    const int W8 = W_ * S_;                    // 2048
    const int H8 = H_ * S_;                    // 1024
    size_t obase = ((size_t)b * H8 + (size_t)h * S_) * W8 + (size_t)t * S_;

    const float4 sv = make_float4(score,  score,  score,  score);
    const float4 lv = make_float4(flabel, flabel, flabel, flabel);
#pragma unroll
    for (int r = 0; r < S_; ++r) {
        size_t o = obase + (size_t)r * W8;     // 32B-aligned (multiple of 8 floats)
        *(float4*)(score_out + o)     = sv;
        *(float4*)(score_out + o + 4) = sv;
        *(float4*)(label_out + o)     = lv;
        *(float4*)(label_out + o + 4) = lv;
    }
}

extern "C" void kernel_launch(void* const* d_in, const int* in_sizes, int n_in,
                              void* d_out, int out_size, void* d_ws, size_t ws_size,
                              hipStream_t stream) {
    (void)in_sizes; (void)n_in; (void)d_ws; (void)ws_size; (void)out_size;

    const float* x = (const float*)d_in[0];                  // [2,19,128,256] f32
    float* score_out = (float*)d_out;                        // first 2*1024*2048 elems
    float* label_out = score_out + (size_t)B_ * (H_ * S_) * (W_ * S_);

    dim3 grid(B_ * H_);   // 256 blocks, one (b,h) row each
    dim3 block(W_);       // 256 threads = 8 waves
    seg_upsample_softmax_argmax<<<grid, block, 0, stream>>>(x, score_out, label_out);
}